// PeakPredictorV1_85023172592363
// MI455X (gfx1250) — compile-verified
//
#include <hip/hip_runtime.h>

// One workgroup (256 threads = 8 wave32) per batch map.
// Memory-bound problem (~277MB traffic -> ~12us floor at 23.3 TB/s); all
// per-batch state staged in LDS (~30KB). Stable sort == JAX argsort via
// 64-bit keys. MLP layers run on V_WMMA_F32_16X16X4_F32 (3 wmma / 16 pos).

#define Hh 23
#define Wd 23
#define Pp 529
#define NPAD 1024
#define PPAD 544
#define NCHUNK 34
#define THR 0.05f

typedef __attribute__((ext_vector_type(2))) float v2f;
typedef __attribute__((ext_vector_type(8))) float v8f;

__device__ __forceinline__ unsigned ordf(float f) {
  unsigned u = __float_as_uint(f);
  return (u & 0x80000000u) ? ~u : (u | 0x80000000u);
}

__device__ __forceinline__ float blockSum(float v, float* red, int tid) {
  red[tid] = v; __syncthreads();
  for (int s = 128; s > 0; s >>= 1) {
    if (tid < s) red[tid] += red[tid + s];
    __syncthreads();
  }
  float r = red[0]; __syncthreads();
  return r;
}

__device__ __forceinline__ unsigned long long
blockMaxU64(unsigned long long v, unsigned long long* red, int tid) {
  red[tid] = v; __syncthreads();
  for (int s = 128; s > 0; s >>= 1) {
    if (tid < s && red[tid + s] > red[tid]) red[tid] = red[tid + s];
    __syncthreads();
  }
  unsigned long long r = red[0]; __syncthreads();
  return r;
}

__device__ __forceinline__ unsigned long long
blockMinU64(unsigned long long v, unsigned long long* red, int tid) {
  red[tid] = v; __syncthreads();
  for (int s = 128; s > 0; s >>= 1) {
    if (tid < s && red[tid + s] < red[tid]) red[tid] = red[tid + s];
    __syncthreads();
  }
  unsigned long long r = red[0]; __syncthreads();
  return r;
}

// Ascending bitonic sort of 1024 u64 keys in LDS, 256 threads.
__device__ __forceinline__ void bitonic1024(unsigned long long* keys, int tid) {
  for (unsigned k = 2; k <= NPAD; k <<= 1) {
    for (unsigned j = k >> 1; j > 0; j >>= 1) {
      __syncthreads();
      for (unsigned t = (unsigned)tid; t < NPAD / 2; t += 256) {
        unsigned i = ((t & ~(j - 1u)) << 1) | (t & (j - 1u));
        unsigned p = i | j;
        unsigned long long a = keys[i], c = keys[p];
        bool up = ((i & k) == 0);
        if ((a > c) == up) { keys[i] = c; keys[p] = a; }
      }
    }
  }
  __syncthreads();
}

__global__ __launch_bounds__(256)
void peak_pred_kernel(const float* __restrict__ ts, const float* __restrict__ anno,
                      const float* __restrict__ W1p, const float* __restrict__ b1p,
                      const float* __restrict__ g1p, const float* __restrict__ be1p,
                      const float* __restrict__ m1p, const float* __restrict__ v1p,
                      const float* __restrict__ W2p, const float* __restrict__ b2p,
                      const float* __restrict__ g2p, const float* __restrict__ be2p,
                      const float* __restrict__ m2p, const float* __restrict__ v2p,
                      const float* __restrict__ W3p, const float* __restrict__ b3p,
                      float* __restrict__ out, int B) {
  __shared__ float sOld[Pp], sCur[Pp], sY[Pp], sTmp[Pp];
  __shared__ unsigned char mOld[Pp], mCur[Pp];
  __shared__ unsigned long long keys[NPAD];
  __shared__ float sErr[PPAD], sSS[PPAD];
  __shared__ float hbuf[8][16][8];           // per-wave 16x8 activation bounce
  __shared__ float redf[256];
  __shared__ unsigned long long redu[256];
  __shared__ float sW1[8][2], sB1[8], sW2[8][8], sB2[8], sW3v[8];
  __shared__ float sB3;

  const int tid = threadIdx.x;
  const int b = blockIdx.x;
  const size_t BP = (size_t)B * Pp;

  // ---- Fold eval-mode BatchNorm into the conv1x1 (linear) weights ----
  if (tid < 8) {
    float a1 = g1p[tid] * rsqrtf(v1p[tid] + 1e-5f);
    sW1[tid][0] = W1p[tid * 2 + 0] * a1;
    sW1[tid][1] = W1p[tid * 2 + 1] * a1;
    sB1[tid] = (b1p[tid] - m1p[tid]) * a1 + be1p[tid];
    float a2 = g2p[tid] * rsqrtf(v2p[tid] + 1e-5f);
#pragma unroll
    for (int k = 0; k < 8; k++) sW2[tid][k] = W2p[tid * 8 + k] * a2;
    sB2[tid] = (b2p[tid] - m2p[tid]) * a2 + be2p[tid];
    sW3v[tid] = W3p[tid];
    if (tid == 0) sB3 = b3p[0];
  }

  // ---- Stage the three 23x23 maps ----
  for (int i = tid; i < Pp; i += 256) {
    sOld[i] = ts[(size_t)b * Pp + i];
    sCur[i] = ts[BP + (size_t)b * Pp + i];
    sY[i]   = anno[(size_t)b * Pp + i];
  }
  __syncthreads();

  // ---- 5x5 max-pool equality masks (separable, SAME w/ -inf pad) ----
  for (int i = tid; i < Pp; i += 256) {
    int c = i % Wd; float m = sOld[i];
    if (c >= 1) m = fmaxf(m, sOld[i - 1]);
    if (c >= 2) m = fmaxf(m, sOld[i - 2]);
    if (c <= Wd - 2) m = fmaxf(m, sOld[i + 1]);
    if (c <= Wd - 3) m = fmaxf(m, sOld[i + 2]);
    sTmp[i] = m;
  }
  __syncthreads();
  for (int i = tid; i < Pp; i += 256) {
    int r = i / Wd; float m = sTmp[i];
    if (r >= 1) m = fmaxf(m, sTmp[i - Wd]);
    if (r >= 2) m = fmaxf(m, sTmp[i - 2 * Wd]);
    if (r <= Hh - 2) m = fmaxf(m, sTmp[i + Wd]);
    if (r <= Hh - 3) m = fmaxf(m, sTmp[i + 2 * Wd]);
    mOld[i] = (sOld[i] == m) && (sOld[i] > THR);
  }
  __syncthreads();
  for (int i = tid; i < Pp; i += 256) {
    int c = i % Wd; float m = sCur[i];
    if (c >= 1) m = fmaxf(m, sCur[i - 1]);
    if (c >= 2) m = fmaxf(m, sCur[i - 2]);
    if (c <= Wd - 2) m = fmaxf(m, sCur[i + 1]);
    if (c <= Wd - 3) m = fmaxf(m, sCur[i + 2]);
    sTmp[i] = m;
  }
  __syncthreads();
  for (int i = tid; i < Pp; i += 256) {
    int r = i / Wd; float m = sTmp[i];
    if (r >= 1) m = fmaxf(m, sTmp[i - Wd]);
    if (r >= 2) m = fmaxf(m, sTmp[i - 2 * Wd]);
    if (r <= Hh - 2) m = fmaxf(m, sTmp[i + Wd]);
    if (r <= Hh - 3) m = fmaxf(m, sTmp[i + 2 * Wd]);
    mCur[i] = (sCur[i] == m) && (sCur[i] > THR);
  }
  __syncthreads();

  // ---- Order-invariant sums + label argmax (deterministic tree reduce) ----
  float pSo = 0.f, pM0o = 0.f, pM1o = 0.f, pSc = 0.f, pM0c = 0.f, pM1c = 0.f;
  unsigned long long aMax = 0ull;
  for (int i = tid; i < Pp; i += 256) {
    float r = (float)(i / Wd), c = (float)(i % Wd);
    if (mOld[i]) { float s = sOld[i]; pSo += s; pM0o += s * r; pM1o += s * c; }
    if (mCur[i]) { float s = sCur[i]; pSc += s; pM0c += s * r; pM1c += s * c; }
    unsigned long long k =
        ((unsigned long long)ordf(sY[i]) << 32) | (unsigned long long)(0xFFFFFFFFu - (unsigned)i);
    if (k > aMax) aMax = k;
  }
  float S_old = blockSum(pSo, redf, tid);
  float M0_old = blockSum(pM0o, redf, tid);
  float M1_old = blockSum(pM1o, redf, tid);
  float S_cur = blockSum(pSc, redf, tid);
  float M0_cur = blockSum(pM0c, redf, tid);
  float M1_cur = blockSum(pM1c, redf, tid);
  unsigned long long mk = blockMaxU64(aMax, redu, tid);
  unsigned gidx = 0xFFFFFFFFu - (unsigned)(mk & 0xFFFFFFFFu);
  float gc0 = (float)(gidx / Wd), gc1 = (float)(gidx % Wd);

  // ---- Sort OLD peaks (key: descending masked score, ties ascending idx) ----
  for (int i = tid; i < NPAD; i += 256) {
    unsigned long long k;
    if (i < Pp) {
      float ms = mOld[i] ? sOld[i] : -__builtin_inff();
      k = ((unsigned long long)(~ordf(ms)) << 32) | (unsigned)i;
    } else {
      k = ~0ull;
    }
    keys[i] = k;
  }
  bitonic1024(keys, tid);

  // nearest valid old peak to gc; tie-break by sorted position
  unsigned long long bestm = ~0ull;
  for (int j = tid; j < NPAD; j += 256) {
    unsigned idx = (unsigned)(keys[j] & 0xFFFFFFFFull);
    float d2 = __builtin_inff();
    if (idx < Pp && mOld[idx]) {
      float dy = gc0 - (float)(idx / Wd);
      float dx = gc1 - (float)(idx % Wd);
      d2 = dy * dy + dx * dx;
    }
    unsigned long long k = ((unsigned long long)ordf(d2) << 32) | (unsigned)j;
    if (k < bestm) bestm = k;
  }
  unsigned long long wk = blockMinU64(bestm, redu, tid);
  unsigned jwin = (unsigned)(wk & 0xFFFFFFFFull);
  unsigned widx = (unsigned)(keys[jwin] & 0xFFFFFFFFull);
  float cg0 = (float)(widx / Wd), cg1 = (float)(widx % Wd);
  float fo0 = S_old * cg0 - M0_old;
  float fo1 = S_old * cg1 - M1_old;
  __syncthreads();

  // ---- Sort CUR peaks ----
  for (int i = tid; i < NPAD; i += 256) {
    unsigned long long k;
    if (i < Pp) {
      float ms = mCur[i] ? sCur[i] : -__builtin_inff();
      k = ((unsigned long long)(~ordf(ms)) << 32) | (unsigned)i;
    } else {
      k = ~0ull;
    }
    keys[i] = k;
  }
  bitonic1024(keys, tid);

  // ---- Sorted outputs + MLP features ----
  for (int j = tid; j < PPAD; j += 256) {
    if (j < Pp) {
      unsigned idx = (unsigned)(keys[j] & 0xFFFFFFFFull);
      float r = (float)(idx / Wd), c = (float)(idx % Wd);
      float sc = sCur[idx];
      bool vv = mCur[idx] != 0;
      size_t base = (size_t)b * Pp + j;
      out[BP + base * 2 + 0] = r;          // c_cur[...,0]
      out[BP + base * 2 + 1] = c;          // c_cur[...,1]
      out[3 * BP + base] = sc;             // s_cur (sorted, unmasked)
      out[4 * BP + base] = vv ? 1.f : 0.f; // val_cur
      float fc0 = S_cur * r - M0_cur;
      float fc1 = S_cur * c - M1_cur;
      float d0 = fo0 - fc0, d1 = fo1 - fc1;
      sErr[j] = sqrtf(fmaxf(d0 * d0 + d1 * d1, 1e-24f));
      sSS[j] = sc;
    } else {
      sErr[j] = 0.f;
      sSS[j] = 0.f;
    }
  }
  __syncthreads();

  // ---- MLP via V_WMMA_F32_16X16X4_F32: 16 positions per chunk per wave ----
  const int lane = tid & 31;
  const int wave = tid >> 5;
  const int n = lane & 15;           // N column (C/D layout) / M row (A layout)
  const bool lo = lane < 16;
  const bool n8 = n < 8;
  const int kb = lo ? 0 : 2;         // A/B K-pair base per lane half
  const int mbase = lo ? 0 : 8;      // C/D row base per lane half

  v2f B1m;                            // B = W1f^T padded 4x16 (K rows 0..1 live)
  B1m[0] = (lo && n8) ? sW1[n][0] : 0.f;
  B1m[1] = (lo && n8) ? sW1[n][1] : 0.f;
  v2f B2a, B2b;                       // B = W2f^T 8x16 split into K 0..3 / 4..7
  B2a[0] = n8 ? sW2[n][kb] : 0.f;
  B2a[1] = n8 ? sW2[n][kb + 1] : 0.f;
  B2b[0] = n8 ? sW2[n][kb + 4] : 0.f;
  B2b[1] = n8 ? sW2[n][kb + 5] : 0.f;
  const float bias1 = n8 ? sB1[n] : 0.f;
  const float bias2 = n8 ? sB2[n] : 0.f;
  float* hb = &hbuf[wave][0][0];

  for (int q = wave; q < NCHUNK; q += 8) {
    const int p0 = q * 16;
    // A = feats 16x4 (K0=err, K1=score, K2=K3=0)
    v2f A1;
    A1[0] = lo ? sErr[p0 + n] : 0.f;
    A1[1] = lo ? sSS[p0 + n] : 0.f;
    v8f C1;
#pragma unroll
    for (int r = 0; r < 8; r++) C1[r] = bias1;
    v8f D1 = __builtin_amdgcn_wmma_f32_16x16x4_f32(false, A1, false, B1m,
                                                   (short)0, C1, false, false);
    if (n8) {
#pragma unroll
      for (int r = 0; r < 8; r++) hb[(mbase + r) * 8 + n] = fmaxf(D1[r], 0.f);
    }
    asm volatile("s_wait_dscnt 0" ::: "memory");
    // A2 = h1 16x8 re-laid (per-lane rows)
    v2f A2a, A2b;
    A2a[0] = hb[n * 8 + kb];     A2a[1] = hb[n * 8 + kb + 1];
    A2b[0] = hb[n * 8 + kb + 4]; A2b[1] = hb[n * 8 + kb + 5];
    v8f C2;
#pragma unroll
    for (int r = 0; r < 8; r++) C2[r] = bias2;
    v8f D2 = __builtin_amdgcn_wmma_f32_16x16x4_f32(false, A2a, false, B2a,
                                                   (short)0, C2, false, false);
    D2 = __builtin_amdgcn_wmma_f32_16x16x4_f32(false, A2b, false, B2b,
                                               (short)0, D2, false, false);
    if (n8) {
#pragma unroll
      for (int r = 0; r < 8; r++) hb[(mbase + r) * 8 + n] = fmaxf(D2[r], 0.f);
    }
    asm volatile("s_wait_dscnt 0" ::: "memory");
    // Layer 3: 16 lanes each finish one position
    if (lo) {
      int pp = p0 + lane;
      if (pp < Pp) {
        float acc = sB3;
#pragma unroll
        for (int t = 0; t < 8; t++) acc += sW3v[t] * hb[lane * 8 + t];
        unsigned idx = (unsigned)(keys[pp] & 0xFFFFFFFFull);
        out[(size_t)b * Pp + pp] = mCur[idx] ? acc : 0.f;
      }
    }
    asm volatile("s_wait_dscnt 0" ::: "memory");
  }
}

extern "C" void kernel_launch(void* const* d_in, const int* in_sizes, int n_in,
                              void* d_out, int out_size, void* d_ws, size_t ws_size,
                              hipStream_t stream) {
  (void)n_in; (void)out_size; (void)d_ws; (void)ws_size;
  const float* ts  = (const float*)d_in[0];
  const float* an  = (const float*)d_in[1];
  const float* W1  = (const float*)d_in[2];
  const float* b1  = (const float*)d_in[3];
  const float* g1  = (const float*)d_in[4];
  const float* be1 = (const float*)d_in[5];
  const float* m1  = (const float*)d_in[6];
  const float* v1  = (const float*)d_in[7];
  const float* W2  = (const float*)d_in[8];
  const float* b2  = (const float*)d_in[9];
  const float* g2  = (const float*)d_in[10];
  const float* be2 = (const float*)d_in[11];
  const float* m2  = (const float*)d_in[12];
  const float* v2  = (const float*)d_in[13];
  const float* W3  = (const float*)d_in[14];
  const float* b3  = (const float*)d_in[15];
  float* out = (float*)d_out;
  int B = in_sizes[0] / (2 * Pp);
  dim3 grid((unsigned)B), block(256);
  hipLaunchKernelGGL(peak_pred_kernel, grid, block, 0, stream,
                     ts, an, W1, b1, g1, be1, m1, v1,
                     W2, b2, g2, be2, m2, v2, W3, b3, out, B);
}